// MultiLayerLogicGateNet_26654567039350
// MI455X (gfx1250) — compile-verified
//
#include <hip/hip_runtime.h>
#include <cmath>
#include <cstdint>
#include <cstddef>

// ---------------------------------------------------------------------------
// MultiLayerLogicGateNet for MI455X (gfx1250, wave32).
// No matmul structure exists (softmax over elementwise products), so WMMA is
// inapplicable; the CDNA5 features used are the async tensor/LDS DMA path
// (global_load_async_to_lds_b32 + s_wait_asynccnt) with double buffering,
// and a v_exp_f32-centric inner loop.
// ---------------------------------------------------------------------------

#define KC         32          // k-chunk per LDS stage
#define LDS_STRIDE 33          // 32 + 1 pad: odd stride -> conflict-free wave32

// CDNA5 async global->LDS copy (ASYNCcnt-tracked DMA, bypasses VGPRs).
__device__ __forceinline__ void async_b32(uint32_t lds_off, const float* g) {
    asm volatile("global_load_async_to_lds_b32 %0, %1, off"
                 :: "v"(lds_off), "v"((uint64_t)(uintptr_t)g)
                 : "memory");
}
__device__ __forceinline__ void wait_async0() {
    asm volatile("s_wait_asynccnt 0x0" ::: "memory");
}
// LDS flat-aperture addresses keep the byte offset in the low 32 bits.
__device__ __forceinline__ uint32_t lds_off32(const void* p) {
    return (uint32_t)(uintptr_t)p;
}

// h0[b, 0:512] = x[b, :], h0[b, 512:1024] = 1 - x[b, :]
__global__ __launch_bounds__(256) void concat_invert_kernel(
    const float* __restrict__ x, float* __restrict__ h0)
{
    int i = blockIdx.x * 256 + threadIdx.x;   // 128*1024 total
    int b = i >> 10;
    int k = i & 1023;
    float v = (k < 512) ? x[b * 512 + k] : 1.0f - x[b * 512 + (k - 512)];
    h0[i] = v;
}

// One or_gate_layer + invert:  hout[b,o] = 1 - sum_k p*z / (p = softmax_k(tau*z))
// Tile: 32 batches x 32 outputs per block (256 threads = 8 wave32).
// Each wave owns 4 output rows (LDS broadcast reads); lane = batch row.
__global__ __launch_bounds__(256) void or_layer_kernel(
    const float* __restrict__ hin,   // (128, in_f)
    const float* __restrict__ W,     // (out_f, in_f) raw weights
    const float* __restrict__ taup,  // scalar tau_adder
    float*       __restrict__ hout,  // (128, out_f)
    int in_f, int out_f, float tau_floor)
{
    __shared__ float xs[2][32 * LDS_STRIDE];   // activation tiles (double buf)
    __shared__ float as[2][32 * LDS_STRIDE];   // clamped-weight tiles

    const int tid  = threadIdx.x;
    const int lane = tid & 31;        // batch within tile
    const int wv   = tid >> 5;        // wave id: owns outputs wv*4 .. wv*4+3
    const int b0   = blockIdx.y << 5;
    const int o0   = blockIdx.x << 5;

    const float t   = *taup;
    const float tau = tau_floor + (t >= 0.0f ? t : 0.05f * t);   // leaky_relu(0.05)
    const float sc  = tau * 1.4426950408889634f;                  // tau * log2(e)

    float se[4] = {0.f, 0.f, 0.f, 0.f};   // sum exp
    float sz[4] = {0.f, 0.f, 0.f, 0.f};   // sum exp*z

    const int nch = in_f >> 5;

    // Each wave DMAs 4 x-rows and 4 W-rows of the chunk (coalesced 128B rows).
    auto issue = [&](int ch, int buf) {
        const int k0 = ch * KC;
        #pragma unroll
        for (int r = 0; r < 4; ++r) {
            const int row = (wv << 2) + r;
            async_b32(lds_off32(&xs[buf][row * LDS_STRIDE + lane]),
                      hin + (size_t)(b0 + row) * in_f + k0 + lane);
            async_b32(lds_off32(&as[buf][row * LDS_STRIDE + lane]),
                      W   + (size_t)(o0 + row) * in_f + k0 + lane);
        }
    };
    // leaky_clamp(W,0,1,0.1) applied once per tile (amortized over 32 batches)
    auto transform = [&](int buf) {
        #pragma unroll
        for (int r = 0; r < 4; ++r) {
            const int idx = ((wv << 2) + r) * LDS_STRIDE + lane;
            float v  = as[buf][idx];
            float hi = fmaf(0.1f, v - 1.0f, 1.0f);
            float v2 = (v > 1.0f) ? hi : v;
            as[buf][idx] = (v < 0.0f) ? 0.1f * v : v2;
        }
    };

    issue(0, 0);
    wait_async0();
    __syncthreads();
    transform(0);
    __syncthreads();

    for (int ch = 0; ch < nch; ++ch) {
        const int cur = ch & 1;
        if (ch + 1 < nch) issue(ch + 1, cur ^ 1);   // DMA overlaps compute

        const float* xrow = &xs[cur][lane * LDS_STRIDE];          // stride 33: no bank conflicts
        const float* a0   = &as[cur][(((wv << 2) + 0)) * LDS_STRIDE];  // wave-uniform: LDS broadcast
        const float* a1   = &as[cur][(((wv << 2) + 1)) * LDS_STRIDE];
        const float* a2   = &as[cur][(((wv << 2) + 2)) * LDS_STRIDE];
        const float* a3   = &as[cur][(((wv << 2) + 3)) * LDS_STRIDE];

        #pragma unroll 4
        for (int k = 0; k < KC; ++k) {
            const float x = xrow[k];
            {   float z = x * a0[k]; float e = __builtin_amdgcn_exp2f(sc * z);
                se[0] += e; sz[0] = fmaf(e, z, sz[0]); }
            {   float z = x * a1[k]; float e = __builtin_amdgcn_exp2f(sc * z);
                se[1] += e; sz[1] = fmaf(e, z, sz[1]); }
            {   float z = x * a2[k]; float e = __builtin_amdgcn_exp2f(sc * z);
                se[2] += e; sz[2] = fmaf(e, z, sz[2]); }
            {   float z = x * a3[k]; float e = __builtin_amdgcn_exp2f(sc * z);
                se[3] += e; sz[3] = fmaf(e, z, sz[3]); }
        }

        wait_async0();        // next chunk's DMA done (issued before compute)
        __syncthreads();      // all waves done reading cur, all tiles landed
        if (ch + 1 < nch) { transform(cur ^ 1); __syncthreads(); }
    }

    #pragma unroll
    for (int j = 0; j < 4; ++j) {
        const int o = o0 + (wv << 2) + j;
        hout[(size_t)(b0 + lane) * out_f + o] = 1.0f - sz[j] / se[j];
    }
}

// ---------------------------------------------------------------------------

static inline float tau_floor_for(int in_f) {
    return (float)(std::log((double)(in_f - 1)) + std::log(0.95) - std::log(0.05));
}

extern "C" void kernel_launch(void* const* d_in, const int* in_sizes, int n_in,
                              void* d_out, int out_size, void* d_ws, size_t ws_size,
                              hipStream_t stream) {
    (void)in_sizes; (void)n_in; (void)out_size; (void)ws_size;

    // setup_inputs() dict order: x, W0, tau0, W1, tau1, W2, tau2, W3, tau3
    const float* x    = (const float*)d_in[0];
    const float* W0   = (const float*)d_in[1];
    const float* tau0 = (const float*)d_in[2];
    const float* W1   = (const float*)d_in[3];
    const float* tau1 = (const float*)d_in[4];
    const float* W2   = (const float*)d_in[5];
    const float* tau2 = (const float*)d_in[6];
    const float* W3   = (const float*)d_in[7];
    const float* tau3 = (const float*)d_in[8];
    float* out = (float*)d_out;

    // Activation ping-pong buffers in workspace (max 128*1024 floats each).
    float* buf0 = (float*)d_ws;
    float* buf1 = buf0 + 128 * 1024;

    // h0 = concat(x, 1-x): (128, 1024)
    concat_invert_kernel<<<512, 256, 0, stream>>>(x, buf0);

    // Layer 0: (128,1024) x W0(1024,1024) -> (128,1024)
    or_layer_kernel<<<dim3(1024 / 32, 128 / 32), 256, 0, stream>>>(
        buf0, W0, tau0, buf1, 1024, 1024, tau_floor_for(1024));
    // Layer 1: (128,1024) x W1(512,1024) -> (128,512)
    or_layer_kernel<<<dim3(512 / 32, 128 / 32), 256, 0, stream>>>(
        buf1, W1, tau1, buf0, 1024, 512, tau_floor_for(1024));
    // Layer 2: (128,512) x W2(256,512) -> (128,256)
    or_layer_kernel<<<dim3(256 / 32, 128 / 32), 256, 0, stream>>>(
        buf0, W2, tau2, buf1, 512, 256, tau_floor_for(512));
    // Layer 3: (128,256) x W3(128,256) -> (128,128) == d_out
    or_layer_kernel<<<dim3(128 / 32, 128 / 32), 256, 0, stream>>>(
        buf1, W3, tau3, out, 256, 128, tau_floor_for(256));
}